// JointGrouping_90623809945817
// MI455X (gfx1250) — compile-verified
//
#include <hip/hip_runtime.h>

typedef __attribute__((ext_vector_type(2))) float v2f;
typedef __attribute__((ext_vector_type(8))) float v8f;

namespace {
constexpr int NG = 5;        // groups
constexpr int NJ = 17;       // joints
constexpr int NC = 256;      // hidden
constexpr int ND = 10;       // D_MAX
constexpr int NTOK = 32 * 4096;
constexpr int NTILES = NTOK / 16;   // 8192 token tiles
constexpr int WAVES = 8;
constexpr int BLK = WAVES * 32;
constexpr int NT = NC / 16;         // 16 column tiles of the 256-wide output
constexpr int WELEMS = NG * NT * 3 * 32;  // 7680 float2 = 60 KB
}

__global__ __launch_bounds__(BLK) void jg_wmma_kernel(
    const float* __restrict__ x,      // (B,T,17,2) f32
    const float* __restrict__ w,      // (5,10,256) f32
    const float* __restrict__ bias,   // (5,256) f32
    float* __restrict__ out) {        // (B,T,5,256) f32
  __shared__ v2f   s_w[WELEMS];       // B fragments, WMMA lane layout
  __shared__ float s_bp[NG * NC];     // bias + spatial PE

  const int tid = threadIdx.x;

  // ---- stage weights in V_WMMA_F32_16X16X4_F32 B-fragment layout ----
  // entry e = ((g*NT + nt)*3 + kt)*32 + lane
  for (int e = tid; e < WELEMS; e += BLK) {
    const int lane = e & 31;
    const int kt   = (e >> 5) % 3;
    const int nt   = (e / 96) & 15;
    const int g    = e / (96 * 16);
    const int n    = nt * 16 + (lane & 15);
    const int kk   = 4 * kt + ((lane >= 16) ? 2 : 0);   // K pair base for this lane
    const float w0 = (kk     < ND) ? w[(g * ND + kk    ) * NC + n] : 0.f;
    const float w1 = (kk + 1 < ND) ? w[(g * ND + kk + 1) * NC + n] : 0.f;
    v2f t; t.x = w0; t.y = w1;
    s_w[e] = t;
  }
  // ---- bias + sinusoidal positional encoding, fused ----
  for (int e = tid; e < NG * NC; e += BLK) {
    const int c = e & (NC - 1);
    const int g = e / NC;
    // div[i] = exp(-(2i)*ln(10000)/256); 2i == (c & ~1)
    const float arg = (float)g * expf((float)(c & ~1) * (-0.0359778920780313f));
    const float pe  = (c & 1) ? cosf(arg) : sinf(arg);
    s_bp[e] = bias[e] + pe;
  }
  __syncthreads();

  const int lane   = tid & 31;
  const int wave   = tid >> 5;
  const int lanelo = lane & 15;
  const bool hi    = lane >= 16;

  // padded gather indices / masks (slot 5 handles the K=10,11 zero pad)
  const int   IDXT[NG][6] = {{1, 2, 3, 0, 0, 0},   {4, 5, 6, 0, 0, 0},
                             {0, 7, 8, 9, 10, 0},  {11, 12, 13, 0, 0, 0},
                             {14, 15, 16, 0, 0, 0}};
  const float MSKT[NG][6] = {{1, 1, 1, 0, 0, 0},   {1, 1, 1, 0, 0, 0},
                             {1, 1, 1, 1, 1, 0},   {1, 1, 1, 0, 0, 0},
                             {1, 1, 1, 0, 0, 0}};
  const int   KTMAX[NG]   = {2, 2, 3, 2, 2};  // len-3 groups: kt=2 fragment is all zero

  const float2* x2 = (const float2*)x;  // one (ch0,ch1) pair per (token,joint)

  for (int tile = blockIdx.x * WAVES + wave; tile < NTILES;
       tile += gridDim.x * WAVES) {
    const int token = tile * 16 + lanelo;  // M row handled by this lane

    // ---- gather A fragments: the gather IS the fragment load ----
    v2f af[NG][3];
#pragma unroll
    for (int g = 0; g < NG; ++g) {
#pragma unroll
      for (int kt = 0; kt < 3; ++kt) {
        if (kt >= KTMAX[g]) continue;
        const int   jj = hi ? IDXT[g][2 * kt + 1] : IDXT[g][2 * kt];
        const float m  = hi ? MSKT[g][2 * kt + 1] : MSKT[g][2 * kt];
        const float2 v = x2[token * NJ + jj];
        v2f a; a.x = v.x * m; a.y = v.y * m;
        af[g][kt] = a;
      }
    }

    // ---- 5 groups x 16 column tiles of 16x16 f32 WMMA ----
#pragma unroll
    for (int g = 0; g < NG; ++g) {
      for (int nt = 0; nt < NT; ++nt) {
        const float bp = s_bp[g * NC + nt * 16 + lanelo];  // bias+PE: col-only
        v8f acc = {bp, bp, bp, bp, bp, bp, bp, bp};
#pragma unroll
        for (int kt = 0; kt < 3; ++kt) {
          if (kt >= KTMAX[g]) continue;
          const v2f b = s_w[((g * NT + nt) * 3 + kt) * 32 + lane];
          acc = __builtin_amdgcn_wmma_f32_16x16x4_f32(
              /*neg_a=*/false, af[g][kt], /*neg_b=*/false, b,
              /*c_mod=*/(short)0, acc, /*reuse_a=*/false, /*reuse_b=*/false);
        }
        // D layout: lane -> column (N = nt*16 + lanelo), VGPR r -> row M
        const int base =
            ((tile * 16 + (hi ? 8 : 0)) * NG + g) * NC + nt * 16 + lanelo;
#pragma unroll
        for (int r = 0; r < 8; ++r) out[base + r * NG * NC] = acc[r];
      }
    }
  }
}

extern "C" void kernel_launch(void* const* d_in, const int* in_sizes, int n_in,
                              void* d_out, int out_size, void* d_ws, size_t ws_size,
                              hipStream_t stream) {
  const float* x    = (const float*)d_in[0];
  const float* w    = (const float*)d_in[1];
  const float* bias = (const float*)d_in[2];
  float* out        = (float*)d_out;
  dim3 grid(NTILES / WAVES);  // 1024 blocks x 8 waves = 8192 tiles, exactly
  dim3 block(BLK);
  jg_wmma_kernel<<<grid, block, 0, stream>>>(x, w, bias, out);
}